// DecouplingFlowLayer_48730698940772
// MI455X (gfx1250) — compile-verified
//
#include <hip/hip_runtime.h>

// DecouplingFlowLayer: Haar-lifting butterfly over the sequence axis.
//   x  : (32, 512, 100, 6) f32, row = N*F = 600 floats = 150 float4
//   X_l[2k] = X_l[2k+1] = (x[2k]+x[2k+1])/2
//   X_h[2k] =  (x[2k]-x[2k+1])/2 ;  X_h[2k+1] = -(x[2k]-x[2k+1])/2
// Memory-bound: ~118 MB total traffic -> ~5us floor at 23.3 TB/s.
// CDNA5 path: async global->LDS staging (ASYNCcnt) + b128 stores, RT caching
// (whole working set fits in the 192MB L2, harness replays the same graph).

#define R4     150               // float4 per row (N*F/4)
#define PAIR4  300               // float4 per even+odd row pair
#define PPB    4                 // row pairs per block
#define TILE4  (PPB * PAIR4)     // 1200 float4 = 19.2 KB LDS
#define NPAIR  8192              // B * S/2 = 32*256
#define L_F4   2457600           // float4 count of the X_l region (32*512*600/4)

__global__ __launch_bounds__(256)
void DecouplingFlowLayer_kernel(const float* __restrict__ xin,
                                float* __restrict__ out) {
  __shared__ float4 tile[TILE4];

  const int tid = threadIdx.x;
  const long long p0 = (long long)blockIdx.x * PPB;   // first pair of this block
  if (p0 >= NPAIR) return;

  // ---- Stage 1: async copy the contiguous input tile into LDS ----------
  // Pair p occupies float4 range [300p, 300p+300) of the flat input, so the
  // block tile is one contiguous 19.2 KB chunk -> fully coalesced b128 loads.
  const float4* __restrict__ src = reinterpret_cast<const float4*>(xin) + p0 * PAIR4;
  // Low 32 bits of the generic shared address are the LDS byte offset.
  const unsigned lds_base = (unsigned)(unsigned long long)(uintptr_t)(&tile[0]);

  for (int i = tid; i < TILE4; i += 256) {
    const unsigned long long ga = (unsigned long long)(uintptr_t)(src + i);
    const unsigned la = lds_base + (unsigned)i * 16u;
    asm volatile("global_load_async_to_lds_b128 %0, %1, off"
                 :: "v"(la), "v"(ga)
                 : "memory");
  }
  // Wait for this wave's async transfers, then cross-wave barrier so every
  // thread can read LDS words written by other waves' transfers.
  asm volatile("s_wait_asynccnt 0x0" ::: "memory");
  __syncthreads();

  // ---- Stage 2: butterfly from LDS, 4x b128 stores per element ----------
  float4* __restrict__ L = reinterpret_cast<float4*>(out);   // X_l region
  float4* __restrict__ H = L + L_F4;                         // X_h region

  for (int j = tid; j < PPB * R4; j += 256) {
    const int pr = j / R4;           // local pair within tile
    const int v  = j - pr * R4;      // float4 index within the row

    const float4 e = tile[pr * PAIR4 + v];        // even row  (ds_load_b128)
    const float4 o = tile[pr * PAIR4 + R4 + v];   // odd row   (ds_load_b128)

    float4 a, d, nd;
    a.x = (e.x + o.x) * 0.5f;  d.x = (e.x - o.x) * 0.5f;  nd.x = -d.x;
    a.y = (e.y + o.y) * 0.5f;  d.y = (e.y - o.y) * 0.5f;  nd.y = -d.y;
    a.z = (e.z + o.z) * 0.5f;  d.z = (e.z - o.z) * 0.5f;  nd.z = -d.z;
    a.w = (e.w + o.w) * 0.5f;  d.w = (e.w - o.w) * 0.5f;  nd.w = -d.w;

    // Global even-row float4 offset: (2*gp)*150 + v == (p0+pr)*300 + v
    const long long re = (p0 + pr) * (long long)PAIR4 + v;
    L[re]      = a;     // X_l even row
    L[re + R4] = a;     // X_l odd row (repeat)
    H[re]      = d;     // X_h even row:  +dif
    H[re + R4] = nd;    // X_h odd row:   -dif
  }
}

extern "C" void kernel_launch(void* const* d_in, const int* in_sizes, int n_in,
                              void* d_out, int out_size, void* d_ws, size_t ws_size,
                              hipStream_t stream) {
  (void)in_sizes; (void)n_in; (void)out_size; (void)d_ws; (void)ws_size;
  const float* x = (const float*)d_in[0];
  float* out = (float*)d_out;
  dim3 grid(NPAIR / PPB);   // 2048 blocks
  dim3 block(256);          // 8 waves (wave32)
  DecouplingFlowLayer_kernel<<<grid, block, 0, stream>>>(x, out);
}